// ModelStep_57715770523995
// MI455X (gfx1250) — compile-verified
//
#include <hip/hip_runtime.h>
#include <math.h>

// Batched Dormand-Prince 5(4) integrator (one lane per ODE system),
// faithful to jax.experimental.ode.odeint semantics in float32.
//
// MI455X notes: workload is per-lane scalar VALU (adaptive ODE stepping with
// divergent trip counts) — wave32 minimizes straggler cost; memory traffic is
// ~5.6MB total (negligible at 23.3 TB/s). The stoichiometric contraction S@v
// has only 0/+-1 entries and is folded into 10 adds; matrix units (WMMA) are
// not applicable to this kernel shape. Hot-path divisions use v_rcp_f32 and
// the step controller uses v_log_f32/v_exp_f32 hardware transcendentals —
// controller precision does not affect solution accuracy (it only picks dt).

namespace {

constexpr float F_ATOL = 1e-6f;
constexpr float F_RTOL = 1e-12f;
constexpr int   MXSTEP = 1000;

__device__ __forceinline__ float fast_rcp(float x)  { return __builtin_amdgcn_rcpf(x); }
__device__ __forceinline__ float fast_sqrt(float x) { return __builtin_amdgcn_sqrtf(x); }
// x^e for x > 0 via hardware log2/exp2 (v_log_f32 / v_exp_f32)
__device__ __forceinline__ float fast_pow(float x, float e) {
  return __builtin_amdgcn_exp2f(e * __builtin_amdgcn_logf(x));
}

struct Params {
  float a1;                         // (c2+c6)/(c2*c6)  (den1 coeff of M)
  float i4, i8;                     // 1/c4, 1/c8       (den1 terms)
  float i10, i12, i14, i16;         // den2 reciprocals
  float K1, K2, K3, K4, K5, K6, K7; // fused rate prefactors cell*k*E/Km
};

__device__ __forceinline__ void eval_rates(const Params& p, const float y[4], float dy[4]) {
  const float M = y[0], MpY = y[1], Mpp = y[2], MpT = y[3];
  const float den1 = 1.0f + M * p.a1 + MpY * p.i4 + MpT * p.i8;
  const float den2 = 1.0f + Mpp * p.i10 + MpT * p.i12 + MpY * p.i14 + M * p.i16;
  const float r1 = fast_rcp(den1);   // den >= ~1 for physical states
  const float r2 = fast_rcp(den2);
  const float v1 = p.K1 * M   * r1;
  const float v2 = p.K2 * MpY * r1;
  const float v3 = p.K3 * M   * r1;
  const float v4 = p.K4 * MpT * r1;
  const float v5 = p.K5 * Mpp * r2;
  const float v6 = p.K6 * MpT * r2;
  const float v7 = p.K7 * MpY * r2;
  // S @ v with S entries in {0,+1,-1}
  dy[0] = v6 + v7 - v1 - v3;
  dy[1] = v1 - v2 - v7;
  dy[2] = v2 + v4 - v5;
  dy[3] = v3 + v5 - v4 - v6;
}

} // namespace

__global__ __launch_bounds__(256) void dopri5_mapk_kernel(
    const float* __restrict__ y_in,
    const float* __restrict__ c_in,
    const float* __restrict__ t_in,
    const float* __restrict__ dT_in,
    float* __restrict__ y_out,
    float* __restrict__ t_out,
    int nsys) {
  const int gid = blockIdx.x * blockDim.x + threadIdx.x;
  if (gid >= nsys) return;

  const float t0     = t_in[0];
  const float target = t0 + dT_in[0];
  if (gid == 0) t_out[0] = target;

  // ---- load per-system constants ----
  float cc[18];
  const float* cp = c_in + (size_t)gid * 18;
#pragma unroll
  for (int j = 0; j < 18; ++j) cc[j] = cp[j];

  const float MEK = cc[0], MKP3 = cc[1], cell = cc[17];
  Params p;
  p.a1  = (cc[2] + cc[6]) / (cc[2] * cc[6]);
  p.i4  = 1.0f / cc[4];
  p.i8  = 1.0f / cc[8];
  p.i10 = 1.0f / cc[10];
  p.i12 = 1.0f / cc[12];
  p.i14 = 1.0f / cc[14];
  p.i16 = 1.0f / cc[16];
  p.K1 = cell * cc[3]  * MEK  / cc[2];
  p.K2 = cell * cc[5]  * MEK  / cc[4];
  p.K3 = cell * cc[7]  * MEK  / cc[6];
  p.K4 = cell * cc[9]  * MEK  / cc[8];
  p.K5 = cell * cc[11] * MKP3 / cc[10];
  p.K6 = cell * cc[13] * MKP3 / cc[12];
  p.K7 = cell * cc[15] * MKP3 / cc[14];

  // ---- load state (coalesced b128) ----
  const float4 y4 = reinterpret_cast<const float4*>(y_in)[gid];
  float y[4] = {y4.x, y4.y, y4.z, y4.w};

  // ---- Dormand-Prince tableau (f32, as in jax.experimental.ode) ----
  const float B21 = (float)(1.0/5.0);
  const float B31 = (float)(3.0/40.0),        B32 = (float)(9.0/40.0);
  const float B41 = (float)(44.0/45.0),       B42 = (float)(-56.0/15.0),
              B43 = (float)(32.0/9.0);
  const float B51 = (float)(19372.0/6561.0),  B52 = (float)(-25360.0/2187.0),
              B53 = (float)(64448.0/6561.0),  B54 = (float)(-212.0/729.0);
  const float B61 = (float)(9017.0/3168.0),   B62 = (float)(-355.0/33.0),
              B63 = (float)(46732.0/5247.0),  B64 = (float)(49.0/176.0),
              B65 = (float)(-5103.0/18656.0);
  const float A1 = (float)(35.0/384.0),  A3 = (float)(500.0/1113.0),
              A4 = (float)(125.0/192.0), A5 = (float)(-2187.0/6784.0),
              A6 = (float)(11.0/84.0);
  const float E1 = (float)(35.0/384.0 - 1951.0/21600.0);
  const float E3 = (float)(500.0/1113.0 - 22642.0/50085.0);
  const float E4 = (float)(125.0/192.0 - 451.0/720.0);
  const float E5 = (float)(-2187.0/6784.0 + 12231.0/42400.0);
  const float E6 = (float)(11.0/84.0 - 649.0/6300.0);
  const float E7 = (float)(-1.0/60.0);
  const float CM1 = (float)(6025192743.0/30085553152.0/2.0);
  const float CM3 = (float)(51252292925.0/65400821598.0/2.0);
  const float CM4 = (float)(-2691868925.0/45128329728.0/2.0);
  const float CM5 = (float)(187940372067.0/1594534317056.0/2.0);
  const float CM6 = (float)(-1776094331.0/19743644256.0/2.0);
  const float CM7 = (float)(11237099.0/235043384.0/2.0);

  // ---- f0 and initial step size (jax initial_step_size, order=4) ----
  float f[4];
  eval_rates(p, y, f);

  float iscale[4];
#pragma unroll
  for (int j = 0; j < 4; ++j) iscale[j] = fast_rcp(F_ATOL + fabsf(y[j]) * F_RTOL);
  float d0 = 0.0f, d1 = 0.0f;
#pragma unroll
  for (int j = 0; j < 4; ++j) {
    const float a = y[j] * iscale[j]; d0 += a * a;
    const float b = f[j] * iscale[j]; d1 += b * b;
  }
  d0 = fast_sqrt(d0); d1 = fast_sqrt(d1);
  const float h0 = ((d0 < 1e-5f) || (d1 < 1e-5f)) ? 1e-6f : 0.01f * d0 * fast_rcp(d1);
  float ytmp[4];
#pragma unroll
  for (int j = 0; j < 4; ++j) ytmp[j] = y[j] + h0 * f[j];
  float f1[4];
  eval_rates(p, ytmp, f1);
  float d2 = 0.0f;
#pragma unroll
  for (int j = 0; j < 4; ++j) {
    const float a = (f1[j] - f[j]) * iscale[j]; d2 += a * a;
  }
  d2 = fast_sqrt(d2) * fast_rcp(h0);
  float h1;
  if ((d1 <= 1e-15f) && (d2 <= 1e-15f)) h1 = fmaxf(1e-6f, h0 * 1e-3f);
  else                                  h1 = fast_pow(0.01f * fast_rcp(fmaxf(d1, d2)), 0.2f);
  float dt = fminf(100.0f * h0, h1);

  // ---- integration state (interp coeffs init to [y0]*5, as in jax) ----
  float t = t0, last_t = t0;
  float pa[4], pb[4], pc[4], pd[4], pe[4];
#pragma unroll
  for (int j = 0; j < 4; ++j) { pa[j] = y[j]; pb[j] = y[j]; pc[j] = y[j]; pd[j] = y[j]; pe[j] = y[j]; }

  // ---- adaptive Dopri5 loop (FSAL) ----
  int it = 0;
  while ((t < target) && (it < MXSTEP) && (dt > 0.0f)) {
    float k2[4], k3[4], k4[4], k5[4], k6[4], k7[4], ynew[4];
    // k1 == f (FSAL)
#pragma unroll
    for (int j = 0; j < 4; ++j) ytmp[j] = y[j] + dt * (B21 * f[j]);
    eval_rates(p, ytmp, k2);
#pragma unroll
    for (int j = 0; j < 4; ++j) ytmp[j] = y[j] + dt * (B31 * f[j] + B32 * k2[j]);
    eval_rates(p, ytmp, k3);
#pragma unroll
    for (int j = 0; j < 4; ++j) ytmp[j] = y[j] + dt * (B41 * f[j] + B42 * k2[j] + B43 * k3[j]);
    eval_rates(p, ytmp, k4);
#pragma unroll
    for (int j = 0; j < 4; ++j)
      ytmp[j] = y[j] + dt * (B51 * f[j] + B52 * k2[j] + B53 * k3[j] + B54 * k4[j]);
    eval_rates(p, ytmp, k5);
#pragma unroll
    for (int j = 0; j < 4; ++j)
      ytmp[j] = y[j] + dt * (B61 * f[j] + B62 * k2[j] + B63 * k3[j] + B64 * k4[j] + B65 * k5[j]);
    eval_rates(p, ytmp, k6);
#pragma unroll
    for (int j = 0; j < 4; ++j)
      ynew[j] = y[j] + dt * (A1 * f[j] + A3 * k3[j] + A4 * k4[j] + A5 * k5[j] + A6 * k6[j]);
    eval_rates(p, ynew, k7);

    // RMS error ratio: sqrt(mean((err/tol)^2))
    float ss = 0.0f;
#pragma unroll
    for (int j = 0; j < 4; ++j) {
      const float yerr = dt * (E1 * f[j] + E3 * k3[j] + E4 * k4[j] + E5 * k5[j] +
                               E6 * k6[j] + E7 * k7[j]);
      const float tol  = F_ATOL + F_RTOL * fmaxf(fabsf(y[j]), fabsf(ynew[j]));
      const float r    = yerr * fast_rcp(tol);
      ss += r * r;
    }
    const float err = fast_sqrt(ss * 0.25f);

    // next dt (jax optimal_step_size: safety=0.9, ifactor=10, dfactor=0.2, order=5)
    const float dfac   = (err < 1.0f) ? 1.0f : 0.2f;
    const float factor = fminf(10.0f, fmaxf(0.9f * fast_pow(err, -0.2f), dfac));
    float ndt          = (err == 0.0f) ? dt * 10.0f : dt * factor;
    ndt = fmaxf(ndt, 0.0f);

    if (err <= 1.0f) {
      // accepted: refresh interpolation polynomial (interp_fit_dopri)
#pragma unroll
      for (int j = 0; j < 4; ++j) {
        const float ymid = y[j] + dt * (CM1 * f[j] + CM3 * k3[j] + CM4 * k4[j] +
                                        CM5 * k5[j] + CM6 * k6[j] + CM7 * k7[j]);
        const float dk0 = dt * f[j];
        const float dk7 = dt * k7[j];
        pa[j] = -2.0f * dk0 + 2.0f * dk7 -  8.0f * y[j] -  8.0f * ynew[j] + 16.0f * ymid;
        pb[j] =  5.0f * dk0 - 3.0f * dk7 + 18.0f * y[j] + 14.0f * ynew[j] - 32.0f * ymid;
        pc[j] = -4.0f * dk0 +        dk7 - 11.0f * y[j] -  5.0f * ynew[j] + 16.0f * ymid;
        pd[j] = dk0;
        pe[j] = y[j];
        y[j] = ynew[j];
        f[j] = k7[j];          // FSAL
      }
      last_t = t;
      t = t + dt;
    }
    dt = ndt;
    ++it;
  }

  // ---- interpolate to the requested output time ----
  const float r = (target - last_t) / (t - last_t);
  float4 out;
  {
    float o[4];
#pragma unroll
    for (int j = 0; j < 4; ++j)
      o[j] = (((pa[j] * r + pb[j]) * r + pc[j]) * r + pd[j]) * r + pe[j];
    out.x = o[0]; out.y = o[1]; out.z = o[2]; out.w = o[3];
  }
  reinterpret_cast<float4*>(y_out)[gid] = out;
}

extern "C" void kernel_launch(void* const* d_in, const int* in_sizes, int n_in,
                              void* d_out, int out_size, void* d_ws, size_t ws_size,
                              hipStream_t stream) {
  // setup_inputs order: y [B,4], w [B,0], c [B,18], t [], deltaT []
  const float* y  = (const float*)d_in[0];
  const float* c  = (const float*)d_in[2];
  const float* t  = (const float*)d_in[3];
  const float* dT = (const float*)d_in[4];

  const int nsys = in_sizes[0] / 4;

  // output tuple flattened: y_new [B*4] | w_new [0] | c [B*18] | t+deltaT [1]
  float* y_out = (float*)d_out;
  float* c_out = y_out + (size_t)nsys * 4;
  float* t_out = c_out + (size_t)nsys * 18;

  // c is returned unchanged: pure pass-through, do it as a DMA copy
  // (graph-capture safe: async d2d on the same stream).
  (void)hipMemcpyAsync(c_out, c, (size_t)nsys * 18 * sizeof(float),
                       hipMemcpyDeviceToDevice, stream);

  const int threads = 256;
  const int blocks  = (nsys + threads - 1) / threads;
  dopri5_mapk_kernel<<<blocks, threads, 0, stream>>>(y, c, t, dT, y_out, t_out, nsys);
}